// GNNLayer_23948737643070
// MI455X (gfx1250) — compile-verified
//
#include <hip/hip_runtime.h>

typedef __attribute__((ext_vector_type(2))) float v2f;
typedef __attribute__((ext_vector_type(8))) float v8f;

// ---------------------------------------------------------------------------
// Degree computation: dis starts at 1.0 (self loop), accumulate in-degree,
// then convert to 1/sqrt(deg) in place.
// ---------------------------------------------------------------------------
__global__ __launch_bounds__(256) void k_fill_one(float* __restrict__ p, int n) {
    int i = blockIdx.x * 256 + threadIdx.x;
    if (i < n) p[i] = 1.0f;
}

__global__ __launch_bounds__(256) void k_deg_accum(const int* __restrict__ dst,
                                                   float* __restrict__ deg, int e) {
    int i = blockIdx.x * 256 + threadIdx.x;
    if (i < e) atomicAdd(&deg[dst[i]], 1.0f);
}

__global__ __launch_bounds__(256) void k_rsqrt(float* __restrict__ p, int n) {
    int i = blockIdx.x * 256 + threadIdx.x;
    if (i < n) p[i] = __frsqrt_rn(p[i]);
}

// ---------------------------------------------------------------------------
// H[n,64] = X[n,din] @ W[din,64] using V_WMMA_F32_16X16X4_F32.
// One wave handles one 16-row tile and all 64 output columns (4 col-tiles).
// A-frag (16x4 f32): lanes 0-15 -> M=lane, K={0,1}; lanes 16-31 -> K={2,3}.
// B-frag (4x16 f32): lanes 0-15 -> N=lane, K={0,1}; lanes 16-31 -> K={2,3}.
// C/D (16x16 f32): vgpr r, lanes 0-15 -> M=r, N=lane; lanes 16-31 -> M=8+r.
// ---------------------------------------------------------------------------
__global__ __launch_bounds__(256) void k_gemm_wmma(const float* __restrict__ X,
                                                   const float* __restrict__ W,
                                                   float* __restrict__ H,
                                                   int ntiles, int din) {
    const int wave = threadIdx.x >> 5;
    const int lane = threadIdx.x & 31;
    const int tile = blockIdx.x * 8 + wave;
    if (tile >= ntiles) return;              // wave-uniform: EXEC stays all-1s

    const int lrow = lane & 15;              // M (for A) / N (for B) within tile
    const int kh   = (lane >> 4) << 1;       // 0 for lanes 0-15, 2 for 16-31

    const float* xrow = X + (size_t)(tile * 16 + lrow) * din;

    v8f acc0 = {}, acc1 = {}, acc2 = {}, acc3 = {};

    for (int k = 0; k < din; k += 4) {
        v2f a;
        a.x = xrow[k + kh];
        a.y = xrow[k + kh + 1];

        const float* w0 = W + (size_t)(k + kh) * 64 + lrow;  // row k+kh
        const float* w1 = w0 + 64;                           // row k+kh+1
        v2f b0, b1, b2, b3;
        b0.x = w0[0];  b0.y = w1[0];
        b1.x = w0[16]; b1.y = w1[16];
        b2.x = w0[32]; b2.y = w1[32];
        b3.x = w0[48]; b3.y = w1[48];

        acc0 = __builtin_amdgcn_wmma_f32_16x16x4_f32(false, a, false, b0, (short)0, acc0, false, false);
        acc1 = __builtin_amdgcn_wmma_f32_16x16x4_f32(false, a, false, b1, (short)0, acc1, false, false);
        acc2 = __builtin_amdgcn_wmma_f32_16x16x4_f32(false, a, false, b2, (short)0, acc2, false, false);
        acc3 = __builtin_amdgcn_wmma_f32_16x16x4_f32(false, a, false, b3, (short)0, acc3, false, false);
    }

    const int mbase = tile * 16 + ((lane >> 4) << 3);
    float* hp = H + (size_t)mbase * 64 + lrow;
#pragma unroll
    for (int r = 0; r < 8; ++r) {
        hp[(size_t)r * 64 +  0] = acc0[r];
        hp[(size_t)r * 64 + 16] = acc1[r];
        hp[(size_t)r * 64 + 32] = acc2[r];
        hp[(size_t)r * 64 + 48] = acc3[r];
    }
}

// ---------------------------------------------------------------------------
// Self-loop term: AGG[i][c] = H[i][c] * dis[i]^2   (dis[i]^2 == 1/deg[i]).
// Fully initializes AGG (required: ws is poisoned).
// ---------------------------------------------------------------------------
__global__ __launch_bounds__(256) void k_selfloop(const float* __restrict__ H,
                                                  const float* __restrict__ dis,
                                                  float* __restrict__ AGG, int n) {
    long long idx = (long long)blockIdx.x * 256 + threadIdx.x;
    if (idx >= (long long)n * 64) return;
    int i = (int)(idx >> 6);
    float d = dis[i];
    AGG[idx] = H[idx] * d * d;
}

// ---------------------------------------------------------------------------
// Edge scatter: one thread per (edge, column).
// 64 consecutive threads share an edge -> coalesced gather & atomic scatter.
// ---------------------------------------------------------------------------
__global__ __launch_bounds__(256) void k_scatter(const float* __restrict__ H,
                                                 const int* __restrict__ src,
                                                 const int* __restrict__ dst,
                                                 const float* __restrict__ dis,
                                                 float* __restrict__ AGG, int e) {
    long long idx = (long long)blockIdx.x * 256 + threadIdx.x;
    long long e64 = (long long)e * 64;
    if (idx >= e64) return;
    int eid = (int)(idx >> 6);
    int c   = (int)(idx & 63);
    int s = src[eid];
    int d = dst[eid];
    float w = dis[s] * dis[d];
    atomicAdd(&AGG[(size_t)d * 64 + c], H[(size_t)s * 64 + c] * w);
}

// ---------------------------------------------------------------------------
// out[i][c] = relu(AGG[i][c] + b[c])   (may run in place)
// ---------------------------------------------------------------------------
__global__ __launch_bounds__(256) void k_bias_relu(const float* __restrict__ AGG,
                                                   const float* __restrict__ b,
                                                   float* __restrict__ out, int n) {
    long long idx = (long long)blockIdx.x * 256 + threadIdx.x;
    if (idx >= (long long)n * 64) return;
    int c = (int)(idx & 63);
    float v = AGG[idx] + b[c];
    out[idx] = v > 0.0f ? v : 0.0f;
}

// ---------------------------------------------------------------------------
// Host-side orchestration
// ---------------------------------------------------------------------------
extern "C" void kernel_launch(void* const* d_in, const int* in_sizes, int n_in,
                              void* d_out, int out_size, void* d_ws, size_t ws_size,
                              hipStream_t stream) {
    const float* x  = (const float*)d_in[0];
    const int*   ei = (const int*)  d_in[1];
    const float* W1 = (const float*)d_in[2];
    const float* b1 = (const float*)d_in[3];
    const float* W3 = (const float*)d_in[4];
    const float* b3 = (const float*)d_in[5];
    const float* W2 = (const float*)d_in[6];
    const float* b2 = (const float*)d_in[7];

    const int n = in_sizes[0] / 128;   // 100000 nodes
    const int e = in_sizes[1] / 2;     // 3.2M edges
    const int* src = ei;
    const int* dst = ei + e;

    float* dis = (float*)d_ws;                    // [n]
    float* A   = dis + n;                         // [n,64] gemm output
    float* B   = A + (size_t)n * 64;              // [n,64] aggregation / next input
    float* out = (float*)d_out;                   // [n,64]

    const int ntiles    = (n + 15) / 16;          // 6250 (exact)
    const int gemm_grid = (ntiles + 7) / 8;
    const long long nc  = (long long)n * 64;
    const int nc_grid   = (int)((nc + 255) / 256);
    const long long ec  = (long long)e * 64;
    const int ec_grid   = (int)((ec + 255) / 256);
    const int n_grid    = (n + 255) / 256;
    const int e_grid    = (e + 255) / 256;

    // degrees -> dis = 1/sqrt(deg)
    k_fill_one <<<n_grid, 256, 0, stream>>>(dis, n);
    k_deg_accum<<<e_grid, 256, 0, stream>>>(dst, dis, e);
    k_rsqrt    <<<n_grid, 256, 0, stream>>>(dis, n);

    // ---- Layer 1: x[n,128] -> B[n,64]
    k_gemm_wmma<<<gemm_grid, 256, 0, stream>>>(x, W1, A, ntiles, 128);
    k_selfloop <<<nc_grid,   256, 0, stream>>>(A, dis, B, n);
    k_scatter  <<<ec_grid,   256, 0, stream>>>(A, src, dst, dis, B, e);
    k_bias_relu<<<nc_grid,   256, 0, stream>>>(B, b1, B, n);

    // ---- Layer 2: B -> B
    k_gemm_wmma<<<gemm_grid, 256, 0, stream>>>(B, W3, A, ntiles, 64);
    k_selfloop <<<nc_grid,   256, 0, stream>>>(A, dis, B, n);
    k_scatter  <<<ec_grid,   256, 0, stream>>>(A, src, dst, dis, B, e);
    k_bias_relu<<<nc_grid,   256, 0, stream>>>(B, b3, B, n);

    // ---- Layer 3: B -> d_out
    k_gemm_wmma<<<gemm_grid, 256, 0, stream>>>(B, W2, A, ntiles, 64);
    k_selfloop <<<nc_grid,   256, 0, stream>>>(A, dis, out, n);
    k_scatter  <<<ec_grid,   256, 0, stream>>>(A, src, dst, dis, out, e);
    k_bias_relu<<<nc_grid,   256, 0, stream>>>(out, b2, out, n);
}